// ChebConv_39479339384912
// MI455X (gfx1250) — compile-verified
//
#include <hip/hip_runtime.h>
#include <hip/hip_bf16.h>

// ---------------------------------------------------------------------------
// ChebConv (K=4) for MI455X / gfx1250
//   T0 = x, T1 = L~ x, Tk = 2 L~ T{k-1} - T{k-2}
//   out = ELU( [T0|T1|T2|T3] (column order c = f*K + k) @ W + b )
//
// Cheb polynomials stored pair-interleaved:
//   P01[row][f] = {T0, T1}   P23[row][f] = {T2, T3}
// so every WMMA A-operand is one contiguous b64 load (ISA A-layout: lanes
// 0-15 hold K=0,1; lanes 16-31 hold K=2,3). Weight is pre-swizzled in LDS so
// every B-operand is one ds_load_b64.
// ---------------------------------------------------------------------------

typedef __attribute__((ext_vector_type(2))) float v2f;
typedef __attribute__((ext_vector_type(8))) float v8f;

#define CHEB_B   4
#define CHEB_N   50000
#define CHEB_F   64
#define CHEB_E   800000
#define CHEB_K   4
#define KDIM     (CHEB_F * CHEB_K)   // 256
#define ROWS     (CHEB_B * CHEB_N)   // 200000

// ---------------------------------------------------------------------------
// P01[i] = {x[i], 0}
// ---------------------------------------------------------------------------
__global__ __launch_bounds__(256) void init_p01(const float* __restrict__ x,
                                                float2* __restrict__ p01,
                                                long n) {
    long i = (long)blockIdx.x * blockDim.x + threadIdx.x;
    if (i < n) p01[i] = make_float2(x[i], 0.0f);
}

// ---------------------------------------------------------------------------
// P23[i] = {-x[i], -T1[i]}   (T1 lives in P01[i].y)
// ---------------------------------------------------------------------------
__global__ __launch_bounds__(256) void init_p23(const float* __restrict__ x,
                                                const float2* __restrict__ p01,
                                                float2* __restrict__ p23,
                                                long n) {
    long i = (long)blockIdx.x * blockDim.x + threadIdx.x;
    if (i < n) p23[i] = make_float2(-x[i], -p01[i].y);
}

// ---------------------------------------------------------------------------
// Scatter SpMM: dst[2*base] += scale * w[e] * src[sstride*base']  where
// base = (b*N + node)*64 + f. 64 consecutive threads handle one (e, b):
// coalesced gather + coalesced global_atomic_add_f32.
// ---------------------------------------------------------------------------
__global__ __launch_bounds__(256) void spmm_scatter(const int* __restrict__ row,
                                                    const int* __restrict__ col,
                                                    const float* __restrict__ w,
                                                    const float* __restrict__ src,
                                                    int sstride,
                                                    float* __restrict__ dst,
                                                    float scale,
                                                    int total_groups) {
    int g = blockIdx.x * 4 + (threadIdx.x >> 6);   // (edge, batch) group
    int f = threadIdx.x & 63;
    if (g >= total_groups) return;
    int e = g >> 2;                 // B == 4
    int b = g & 3;
    int r = row[e];
    int c = col[e];
    float ww = w[e] * scale;
    float v = src[(((long)b * CHEB_N + c) * CHEB_F + f) * sstride] * ww;
    unsafeAtomicAdd(&dst[(((long)b * CHEB_N + r) * CHEB_F + f) * 2], v);
}

// ---------------------------------------------------------------------------
// Dense GEMM + bias + ELU via V_WMMA_F32_16X16X4_F32.
// Each wave: 16 rows x 64 cols (4 v8f accumulators), 8 waves/block.
// A operand: v2f from P01 (lanes 0-15) / P23 (lanes 16-31).
// B operand: v2f from LDS-swizzled weight: wlds[(c>>1)*128 + col*2 + (c&1)].
// ---------------------------------------------------------------------------
__global__ __launch_bounds__(256) void cheb_gemm_elu(const float* __restrict__ p01,
                                                     const float* __restrict__ p23,
                                                     const float* __restrict__ weight,
                                                     const float* __restrict__ bias,
                                                     float* __restrict__ out,
                                                     int rows) {
    extern __shared__ float wlds[];                    // 64 KB, pair-swizzled

    // cooperative swizzled weight load: pair rows (2c, 2c+1) -> one float2
    for (int i = threadIdx.x; i < (KDIM / 2) * CHEB_F; i += 256) {
        int cpair = i >> 6;                            // 0..127
        int colx  = i & 63;
        float w0 = weight[(cpair * 2 + 0) * CHEB_F + colx];
        float w1 = weight[(cpair * 2 + 1) * CHEB_F + colx];
        ((float2*)wlds)[cpair * CHEB_F + colx] = make_float2(w0, w1);
    }
    __syncthreads();

    const int wave  = threadIdx.x >> 5;                // 0..7
    const int lane  = threadIdx.x & 31;
    const int lhalf = lane >> 4;                       // 0: K=0,1 | 1: K=2,3
    const int lmod  = lane & 15;

    const long m0 = ((long)blockIdx.x * 8 + wave) * 16;
    if (m0 >= rows) return;                            // wave-uniform guard

    const v2f* ap = (const v2f*)((lhalf ? p23 : p01) + (m0 + lmod) * (2 * CHEB_F));
    const v2f* wp = (const v2f*)wlds;

    v8f acc[4] = {v8f{}, v8f{}, v8f{}, v8f{}};

#pragma unroll 4
    for (int f = 0; f < CHEB_F; ++f) {
        const v2f a = ap[f];                           // {T_even, T_odd} one b64
        const v2f* wrow = wp + (f * 2 + lhalf) * CHEB_F;
#pragma unroll
        for (int t = 0; t < 4; ++t) {
            const v2f b = wrow[t * 16 + lmod];         // one ds_load_b64
            acc[t] = __builtin_amdgcn_wmma_f32_16x16x4_f32(
                /*neg_a=*/false, a, /*neg_b=*/false, b,
                /*c_mod=*/(short)0, acc[t],
                /*reuse_a=*/false, /*reuse_b=*/false);
        }
    }

    // C/D layout: VGPR i -> row m0 + i + 8*lhalf, col = t*16 + lmod
#pragma unroll
    for (int t = 0; t < 4; ++t) {
        const int c = t * 16 + lmod;
        const float bi = bias[c];
#pragma unroll
        for (int i = 0; i < 8; ++i) {
            const long r = m0 + i + 8 * lhalf;
            float v = acc[t][i] + bi;
            v = (v > 0.0f) ? v : (__expf(v) - 1.0f);   // ELU
            out[r * (long)CHEB_F + c] = v;
        }
    }
}

// ---------------------------------------------------------------------------
extern "C" void kernel_launch(void* const* d_in, const int* in_sizes, int n_in,
                              void* d_out, int out_size, void* d_ws, size_t ws_size,
                              hipStream_t stream) {
    const float* x      = (const float*)d_in[0];   // [B,N,64]
    const int*   row    = (const int*)d_in[1];     // [E]
    const int*   col    = (const int*)d_in[2];     // [E]
    const float* w      = (const float*)d_in[3];   // [E]
    const float* weight = (const float*)d_in[4];   // [256,64]
    const float* bias   = (const float*)d_in[5];   // [64]
    float*       out    = (float*)d_out;           // [B,N,64]

    const long n = (long)ROWS * CHEB_F;            // 12.8M elements
    float* P01 = (float*)d_ws;                     // [ROWS][64][2]
    float* P23 = P01 + n * 2;                      // [ROWS][64][2]

    dim3 blk(256);
    const int initGrid = (int)((n + 255) / 256);
    const int total_groups = CHEB_E * CHEB_B;      // 3.2M (e,b) pairs
    const int scatGrid = (total_groups + 3) / 4;

    // P01 = {x, 0};  T1 = L~ x accumulated into slot 1
    init_p01<<<initGrid, blk, 0, stream>>>(x, (float2*)P01, n);
    spmm_scatter<<<scatGrid, blk, 0, stream>>>(row, col, w, x, 1, P01 + 1, 1.0f, total_groups);
    // P23 = {-x, -T1};  T2 = 2 L~ T1 - x into slot 0;  T3 = 2 L~ T2 - T1 into slot 1
    init_p23<<<initGrid, blk, 0, stream>>>(x, (const float2*)P01, (float2*)P23, n);
    spmm_scatter<<<scatGrid, blk, 0, stream>>>(row, col, w, P01 + 1, 2, P23 + 0, 2.0f, total_groups);
    spmm_scatter<<<scatGrid, blk, 0, stream>>>(row, col, w, P23 + 0, 2, P23 + 1, 2.0f, total_groups);

    // GEMM + bias + ELU (WMMA f32 16x16x4)
    const int gemmGrid = (ROWS + 127) / 128;       // 128 rows / block
    cheb_gemm_elu<<<gemmGrid, blk, (KDIM / 2) * CHEB_F * sizeof(float2), stream>>>(
        P01, P23, weight, bias, out, ROWS);
}